// ProteinBackboneFeature_35588099015202
// MI455X (gfx1250) — compile-verified
//
#include <hip/hip_runtime.h>
#include <math.h>

// Problem shape (from reference)
#define NB      50                 // beads
#define NPAD    64                 // padded beads for 4x4 WMMA tiling
#define P_DIST  1225               // 50*49/2 pairwise distances
#define N_ANG   48                 // n-2 angles
#define N_DIH   47                 // n-3 dihedrals
#define ROW_LEN 1367               // 1225 + 48 + 47 + 47
#define OFF_ANG  1225
#define OFF_DCOS 1273
#define OFF_DSIN 1320

typedef __attribute__((ext_vector_type(2))) float v2f;
typedef __attribute__((ext_vector_type(8))) float v8f;

__global__ __launch_bounds__(32)
void backbone_feat_kernel(const float* __restrict__ data,
                          float* __restrict__ out)
{
    // SoA coords, zero-padded to 64 beads: s_c[comp*NPAD + bead]
    __shared__ float s_c[3 * NPAD];
    __shared__ float s_n2[NPAD];
    __shared__ float s_row[ROW_LEN + 1];   // [ROW_LEN] = dump slot for masked lanes

    const int f    = blockIdx.x;
    const int lane = threadIdx.x;          // wave32: 0..31

    // ---- zero pad region, then load the frame's 150 floats (coalesced) ----
    for (int t = lane; t < 3 * NPAD; t += 32) s_c[t] = 0.0f;
    __syncthreads();

    const float* src = data + (size_t)f * (NB * 3);
    for (int t = lane; t < NB * 3; t += 32) {
        int b = t / 3;
        int c = t - 3 * b;
        s_c[c * NPAD + b] = src[t];
    }
    __syncthreads();

    // ---- squared norms per bead (pad rows are exactly 0) ----
    for (int b = lane; b < NPAD; b += 32) {
        float x = s_c[b], y = s_c[NPAD + b], z = s_c[2 * NPAD + b];
        s_n2[b] = x * x + y * y + z * z;
    }
    __syncthreads();

    // ---- Gram matrix G = X * X^T via V_WMMA_F32_16X16X4_F32 ----
    // A (16x4, MxK): lanes 0-15 hold K={0,1} in vgpr{0,1}; lanes 16-31 hold K={2,3}.
    // B (4x16, KxN) mirrors it with N in place of M -> identical register build.
    const int half = lane >> 4;
    const int mrow = lane & 15;

    v2f op[4];
#pragma unroll
    for (int r = 0; r < 4; ++r) {
        int row = r * 16 + mrow;
        float e0, e1;
        if (half == 0) { e0 = s_c[row];            e1 = s_c[NPAD + row]; }
        else           { e0 = s_c[2 * NPAD + row]; e1 = 0.0f;            }
        op[r].x = e0;
        op[r].y = e1;
    }

    // Hoist all needed squared norms into registers (no LDS reads in tile loop).
    // i-side: 8 consecutive values per row-block (vectorizes to ds_load_b128 x2).
    // j-side: one value per col-block.
    float n2i[4][8];
    float n2j[4];
#pragma unroll
    for (int r = 0; r < 4; ++r) {
        const int base = r * 16 + 8 * half;
#pragma unroll
        for (int v = 0; v < 8; ++v) n2i[r][v] = s_n2[base + v];
        n2j[r] = s_n2[r * 16 + mrow];
    }

    // Upper-triangle tiles (r <= c): 4 diagonal + 6 off-diagonal = 10 WMMAs.
    // Validity guards specialized per tile:
    //   NEED_IJ: diagonal tiles need i<j ; NEED_JB: c==3 tiles need j<NB.
    const int  TR[10]      = {0, 0, 0, 0, 1, 1, 1, 2, 2, 3};
    const int  TC[10]      = {0, 1, 2, 3, 1, 2, 3, 2, 3, 3};
    const bool NEED_IJ[10] = {1, 0, 0, 0, 1, 0, 0, 1, 0, 1};
    const bool NEED_JB[10] = {0, 0, 0, 1, 0, 0, 1, 0, 1, 1};

#pragma unroll
    for (int t = 0; t < 10; ++t) {
        const int r = TR[t], c = TC[t];
        v8f acc = {};
        // 8 args: (neg_a, A, neg_b, B, c_mod, C, reuse_a, reuse_b)
        acc = __builtin_amdgcn_wmma_f32_16x16x4_f32(
                  false, op[r], false, op[c],
                  (short)0, acc, false, false);

        const int j   = c * 16 + mrow;
        const bool jb = !NEED_JB[t] || (j < NB);

        // C/D layout: vgpr v, lane -> G[i][j], i = 16r + 8*half + v, j = 16c + mrow
#pragma unroll
        for (int v = 0; v < 8; ++v) {
            const int i = r * 16 + 8 * half + v;
            float d2   = n2i[r][v] + n2j[c] - 2.0f * acc[v];
            float dist = __builtin_amdgcn_sqrtf(fmaxf(d2, 0.0f));
            // output ordering: pairs grouped by separation d = j-i
            int dd  = j - i;
            int idx = (dd - 1) * NB - (((dd - 1) * dd) >> 1) + i;
            bool valid = jb && (!NEED_IJ[t] || (i < j));
            idx = valid ? idx : ROW_LEN;        // masked lanes write dump slot
            s_row[idx] = dist;                  // unconditional ds_store
        }
    }

    // ---- angles: vertex at bead m+1 ----
    for (int m = lane; m < N_ANG; m += 32) {
        float ax = s_c[m]            - s_c[m + 1];
        float ay = s_c[NPAD + m]     - s_c[NPAD + m + 1];
        float az = s_c[2*NPAD + m]   - s_c[2*NPAD + m + 1];
        float bx = s_c[m + 2]          - s_c[m + 1];
        float by = s_c[NPAD + m + 2]   - s_c[NPAD + m + 1];
        float bz = s_c[2*NPAD + m + 2] - s_c[2*NPAD + m + 1];
        float dp  = ax * bx + ay * by + az * bz;
        float ct  = dp * __builtin_amdgcn_rsqf(ax*ax + ay*ay + az*az)
                       * __builtin_amdgcn_rsqf(bx*bx + by*by + bz*bz);
        ct = fminf(1.0f, fmaxf(-1.0f, ct));
        s_row[OFF_ANG + m] = acosf(ct);
    }

    // ---- dihedrals over beads (m, m+1, m+2, m+3) ----
    for (int m = lane; m < N_DIH; m += 32) {
        float x0 = s_c[m],     y0 = s_c[NPAD + m],     z0 = s_c[2*NPAD + m];
        float x1 = s_c[m + 1], y1 = s_c[NPAD + m + 1], z1 = s_c[2*NPAD + m + 1];
        float x2 = s_c[m + 2], y2 = s_c[NPAD + m + 2], z2 = s_c[2*NPAD + m + 2];
        float x3 = s_c[m + 3], y3 = s_c[NPAD + m + 3], z3 = s_c[2*NPAD + m + 3];

        // b1 = p0-p1, o1 = p2-p1, b2 = p1-p2 = -o1, o2 = p3-p2
        float b1x = x0 - x1, b1y = y0 - y1, b1z = z0 - z1;
        float o1x = x2 - x1, o1y = y2 - y1, o1z = z2 - z1;
        float b2x = -o1x,    b2y = -o1y,    b2z = -o1z;
        float o2x = x3 - x2, o2y = y3 - y2, o2z = z3 - z2;

        // cp1 = b1 x o1 ; cp2 = b2 x o2 ; pv2 = cp2 x o2
        float c1x = b1y * o1z - b1z * o1y;
        float c1y = b1z * o1x - b1x * o1z;
        float c1z = b1x * o1y - b1y * o1x;
        float c2x = b2y * o2z - b2z * o2y;
        float c2y = b2z * o2x - b2x * o2z;
        float c2z = b2x * o2y - b2y * o2x;
        float pvx = c2y * o2z - c2z * o2y;
        float pvy = c2z * o2x - c2x * o2z;
        float pvz = c2x * o2y - c2y * o2x;

        float r1 = __builtin_amdgcn_rsqf(c1x*c1x + c1y*c1y + c1z*c1z);
        float cosd = (c1x*c2x + c1y*c2y + c1z*c2z) * r1
                     * __builtin_amdgcn_rsqf(c2x*c2x + c2y*c2y + c2z*c2z);
        float sind = (c1x*pvx + c1y*pvy + c1z*pvz) * r1
                     * __builtin_amdgcn_rsqf(pvx*pvx + pvy*pvy + pvz*pvz);

        s_row[OFF_DCOS + m] = cosd;
        s_row[OFF_DSIN + m] = sind;
    }

    __syncthreads();

    // ---- coalesced streaming store of the staged row ----
    // (row base is only 4B-aligned since 1367 is odd -> b32 stores; lanes
    //  still coalesce into full cachelines)
    float* dst = out + (size_t)f * ROW_LEN;
    for (int t = lane; t < ROW_LEN; t += 32)
        dst[t] = s_row[t];
}

extern "C" void kernel_launch(void* const* d_in, const int* in_sizes, int n_in,
                              void* d_out, int out_size, void* d_ws, size_t ws_size,
                              hipStream_t stream)
{
    (void)n_in; (void)out_size; (void)d_ws; (void)ws_size;
    const float* data = (const float*)d_in[0];
    float* out = (float*)d_out;
    int frames = in_sizes[0] / (NB * 3);   // 50000
    backbone_feat_kernel<<<frames, 32, 0, stream>>>(data, out);
}